// PolymorphicNeuron_87222195847625
// MI455X (gfx1250) — compile-verified
//
#include <hip/hip_runtime.h>
#include <hip/hip_bf16.h>

typedef __attribute__((ext_vector_type(2))) float v2f;
typedef __attribute__((ext_vector_type(4))) float v4f;
typedef __attribute__((ext_vector_type(8))) float v8f;

#define B_SZ   1024
#define D_SZ   512
#define M_SZ   256
#define BD_SZ  (B_SZ * D_SZ)   // 524288

// ---------------------------------------------------------------------------
// Kernel 1: logits = x @ W^T + bias using V_WMMA_F32_16X16X4_F32.
// One wave (32 threads) per 16x16 output tile. K = 512 in steps of 4.
// A (16x4 f32): lanes 0-15 -> M=lane, K={0,1}; lanes 16-31 -> M=lane-16, K={2,3}
// B (4x16 f32): lanes 0-15 -> N=lane, K={0,1}; lanes 16-31 -> N=lane-16, K={2,3}
// C (16x16 f32, 8 VGPR): lane half selects M base (+0 / +8), vgpr v -> M=v.
// ---------------------------------------------------------------------------
__global__ void __launch_bounds__(32)
logits_wmma_f32(const float* __restrict__ x, const float* __restrict__ W,
                const float* __restrict__ bias, float* __restrict__ logits) {
    const int rb   = blockIdx.x * 16;      // batch-row tile base
    const int nb   = blockIdx.y * 16;      // mode-col tile base
    const int lane = threadIdx.x;          // 0..31
    const int half = lane >> 4;            // 0: K 0,1   1: K 2,3
    const int l16  = lane & 15;

    const int row  = rb + l16;             // A row (batch index)
    const int col  = nb + l16;             // B col (mode index)
    const int koff = half * 2;

    const float* xa = x + (size_t)row * D_SZ + koff;
    const float* wb = W + (size_t)col * D_SZ + koff;   // W[n][k] == (W^T)[k][n]

    v8f c = {};
    #pragma unroll 8
    for (int k = 0; k < D_SZ; k += 4) {
        v2f a  = *(const v2f*)(xa + k);
        v2f bm = *(const v2f*)(wb + k);
        // 8 args: (neg_a, A, neg_b, B, c_mod, C, reuse_a, reuse_b)
        c = __builtin_amdgcn_wmma_f32_16x16x4_f32(
                false, a, false, bm, (short)0, c, false, false);
    }

    const float bv = bias[col];
    #pragma unroll
    for (int v = 0; v < 8; ++v) {
        const int r = rb + v + half * 8;
        logits[(size_t)r * M_SZ + col] = c[v] + bv;
    }
}

// ---------------------------------------------------------------------------
// Kernel 2: in-place row softmax over M_SZ=256 modes.
// One 256-thread block (8 wave32) per batch row. Each thread owns exactly one
// element, so in-place read->reduce->write has no cross-thread hazards.
// ---------------------------------------------------------------------------
__global__ void __launch_bounds__(256)
softmax256_inplace(float* __restrict__ s) {
    __shared__ float red[8];
    const int b    = blockIdx.x;
    const int t    = threadIdx.x;
    const int wave = t >> 5;
    const int lane = t & 31;

    const float v = s[(size_t)b * M_SZ + t];

    // row max
    float mx = v;
    #pragma unroll
    for (int off = 16; off; off >>= 1) mx = fmaxf(mx, __shfl_xor(mx, off));
    if (lane == 0) red[wave] = mx;
    __syncthreads();
    float rowmax = red[0];
    #pragma unroll
    for (int w = 1; w < 8; ++w) rowmax = fmaxf(rowmax, red[w]);
    __syncthreads();

    // exp + row sum
    const float e = expf(v - rowmax);
    float sm = e;
    #pragma unroll
    for (int off = 16; off; off >>= 1) sm += __shfl_xor(sm, off);
    if (lane == 0) red[wave] = sm;
    __syncthreads();
    float rowsum = 0.f;
    #pragma unroll
    for (int w = 0; w < 8; ++w) rowsum += red[w];

    s[(size_t)b * M_SZ + t] = e / rowsum;
}

// ---------------------------------------------------------------------------
// Kernel 3: the HBM-bound pass, vectorized float4 (b128 per lane).
// One thread per 4 consecutive (b,d) elements; loop over 256 modes.
//   mem_new = beta_m*mem + x - (mem>1)
//   spk     = (mem_new > 1)
//   mixed  += spk * scores[b,m]
// mem/mem_new are single-touch 512 MiB streams -> non-temporal b128 accesses
// so the 192 MB L2 keeps scores/x instead of the dead stream, and each VMEM
// instruction moves a full 256B per half-wave.
// ---------------------------------------------------------------------------
__global__ void __launch_bounds__(256)
leaky_stream4(const float* __restrict__ x, const float* __restrict__ mem,
              const float* __restrict__ scores,
              float* __restrict__ mixed, float* __restrict__ mem_new) {
    const int v    = blockIdx.x * blockDim.x + threadIdx.x;  // 0 .. BD/4-1
    const int e    = v * 4;                                  // element base
    const int b    = e >> 9;                                 // e / D_SZ (4 | 512)
    const v4f  xv  = *(const v4f*)(x + e);
    const float* sc = scores + (size_t)b * M_SZ;

    const float dbeta = 1.0f / (float)(M_SZ + 1);
    v4f acc = {0.f, 0.f, 0.f, 0.f};

    #pragma unroll 4
    for (int m = 0; m < M_SZ; ++m) {
        const size_t off  = (size_t)m * BD_SZ + (size_t)e;
        const v4f    mm   = __builtin_nontemporal_load((const v4f*)(mem + off));
        const float  beta = (float)m * dbeta;
        const float  s    = sc[m];

        v4f mnew;
        #pragma unroll
        for (int i = 0; i < 4; ++i) {
            const float reset = (mm[i] > 1.0f) ? 1.0f : 0.0f;   // from *previous* mem
            mnew[i] = fmaf(beta, mm[i], xv[i]) - reset;
            if (mnew[i] > 1.0f) acc[i] += s;                    // spk * score
        }
        __builtin_nontemporal_store(mnew, (v4f*)(mem_new + off));
    }
    *(v4f*)(mixed + e) = acc;
}

// ---------------------------------------------------------------------------
// Launcher. Inputs: x[B,D], mem[M,B,D], W[M,D], b[M].
// Output layout (flat, return order): mixed[B,D] | mem_new[M,B,D] | scores[B,M]
// ---------------------------------------------------------------------------
extern "C" void kernel_launch(void* const* d_in, const int* in_sizes, int n_in,
                              void* d_out, int out_size, void* d_ws, size_t ws_size,
                              hipStream_t stream) {
    const float* x    = (const float*)d_in[0];
    const float* mem  = (const float*)d_in[1];
    const float* W    = (const float*)d_in[2];
    const float* bias = (const float*)d_in[3];

    float* out     = (float*)d_out;
    float* mixed   = out;
    float* mem_new = out + (size_t)BD_SZ;
    float* scores  = out + (size_t)BD_SZ + (size_t)M_SZ * BD_SZ;

    // 1) logits -> scores slice (in-place softmax next)
    logits_wmma_f32<<<dim3(B_SZ / 16, M_SZ / 16), 32, 0, stream>>>(x, W, bias, scores);
    // 2) softmax rows in place
    softmax256_inplace<<<B_SZ, 256, 0, stream>>>(scores);
    // 3) fused leak/reset/spike/mix stream (float4 per thread)
    leaky_stream4<<<(BD_SZ / 4) / 256, 256, 0, stream>>>(x, mem, scores, mixed, mem_new);
}